// E3nnMLPNorm_63024350102049
// MI455X (gfx1250) — compile-verified
//
#include <hip/hip_runtime.h>
#include <math.h>

#define MULC 1024
#define BSZ  1024
#define BV   (BSZ * MULC)

typedef __attribute__((ext_vector_type(2))) float v2f;
typedef __attribute__((ext_vector_type(8))) float v8f;

// ---------------------------------------------------------------------------
// CDNA5 async global->LDS path (ASYNCcnt-tracked), with sync fallback
// ---------------------------------------------------------------------------
#if __has_builtin(__builtin_amdgcn_global_load_async_to_lds_b128) && \
    __has_builtin(__builtin_amdgcn_s_wait_asynccnt)
#define USE_ASYNC 1
#else
#define USE_ASYNC 0
#endif

#if USE_ASYNC
// builtin expects pointers to 16-byte int vectors (per hipcc diagnostic)
typedef int vi4 __attribute__((vector_size(16)));
typedef __attribute__((address_space(1))) vi4* gas1_p;
typedef __attribute__((address_space(3))) vi4* lds3_p;
// Generic LDS addresses carry the LDS byte offset in the low 32 bits
// AS1 representation == generic.
__device__ __forceinline__ void async_cp16(const float* g, float* l) {
    __builtin_amdgcn_global_load_async_to_lds_b128(
        (gas1_p)(unsigned long long)(const void*)g,
        (lds3_p)(unsigned)(unsigned long long)(void*)l,
        0, 0);
}
#endif

// ---------------------------------------------------------------------------
// Block 1: f[i][b][v] = (x[b,0,i]*w1[0,v] + x[b,1,i]*w1[1,v]) / sqrt(2)
// field buffers layout: [3][B][MUL], v fastest
// ---------------------------------------------------------------------------
__global__ __launch_bounds__(256) void block1_kernel(const float* __restrict__ x,
                                                     const float* __restrict__ w1,
                                                     float* __restrict__ A) {
    int idx = blockIdx.x * 256 + threadIdx.x;     // (i*B + b)*MUL + v
    int v  = idx & (MULC - 1);
    int ib = idx >> 10;
    int b  = ib & (BSZ - 1);
    int i  = ib >> 10;
    const float inv_sqrt2 = 0.70710678118654752440f;
    float x0 = x[(b * 2 + 0) * 3 + i];
    float x1 = x[(b * 2 + 1) * 3 + i];
    A[idx] = (x0 * w1[v] + x1 * w1[MULC + v]) * inv_sqrt2;
}

// ---------------------------------------------------------------------------
// var[v] = mean over (i,b) of f^2  (3072 rows per channel)
// ---------------------------------------------------------------------------
__global__ __launch_bounds__(256) void var_zero(float* __restrict__ var) {
    var[blockIdx.x * 256 + threadIdx.x] = 0.0f;
}

__global__ __launch_bounds__(256) void var_reduce(const float* __restrict__ A,
                                                  float* __restrict__ var) {
    int v  = blockIdx.x * 256 + threadIdx.x;
    int r0 = blockIdx.y * 192;
    float acc = 0.0f;
    for (int r = r0; r < r0 + 192; ++r) {
        float f = A[(size_t)r * MULC + v];
        acc += f * f;
    }
    atomicAdd(&var[v], acc * (1.0f / 3072.0f));
}

// ---------------------------------------------------------------------------
// bn + norm-gated sigmoid
// ---------------------------------------------------------------------------
__global__ __launch_bounds__(256) void bn_act_kernel(const float* __restrict__ A,
                                                     const float* __restrict__ var,
                                                     const float* __restrict__ w,
                                                     float* __restrict__ Out) {
    int idx = blockIdx.x * 256 + threadIdx.x;     // b*MUL + v
    int v = idx & (MULC - 1);
    float rs = rsqrtf(var[v] + 1e-5f) * w[v];
    float f0 = A[idx]          * rs;
    float f1 = A[BV + idx]     * rs;
    float f2 = A[2 * BV + idx] * rs;
    float n  = sqrtf(f0 * f0 + f1 * f1 + f2 * f2 + 1e-8f);
    float g  = 1.0f / ((1.0f + expf(-n)) * n);    // sigmoid(n)/n
    Out[idx]          = f0 * g;
    Out[BV + idx]     = f1 * g;
    Out[2 * BV + idx] = f2 * g;
}

// ---------------------------------------------------------------------------
// GEMM: G[z][b][v] = scale * sum_u F[z][b][u] * W[u][v]    (1024x1024x1024)
// 128x128 tile / workgroup, 8 waves, 64x32 per wave (4x2 WMMA tiles),
// V_WMMA_F32_16X16X4_F32; K staged 32-wide through double-buffered LDS via
// global_load_async_to_lds_b128 (ASYNCcnt pipeline).
// ---------------------------------------------------------------------------
#define KT 32
#define ASTR 36          // padded A row stride (144B: 16B aligned, bank-spread)
#define NKT (MULC / KT)  // 32 k-tiles

__global__ __launch_bounds__(256) void gemm_wmma_kernel(const float* __restrict__ Fin,
                                                        const float* __restrict__ Wt,
                                                        float* __restrict__ Gout,
                                                        float scale) {
    __shared__ float As[2][128 * ASTR];   // 128 rows x 32 k (padded)
    __shared__ float Bs[2][KT * 128];     // 32 k x 128 n

    const int tid  = threadIdx.x;
    const int lane = tid & 31;
    const int wave = tid >> 5;
    const int wm   = wave >> 2;      // 0..1  (64-row slab)
    const int wn   = wave & 3;       // 0..3  (32-col slab)
    const int bm   = blockIdx.x;
    const int bn   = blockIdx.y;

    const float* Fz = Fin + (size_t)blockIdx.z * BV;
    float*       Gz = Gout + (size_t)blockIdx.z * BV;

    v8f acc[4][2];
    const v8f vzero = {0.f, 0.f, 0.f, 0.f, 0.f, 0.f, 0.f, 0.f};
#pragma unroll
    for (int mt = 0; mt < 4; ++mt)
#pragma unroll
        for (int nt = 0; nt < 2; ++nt) acc[mt][nt] = vzero;

    const int ml = lane & 15;        // row (A) / col (B) within 16
    const int kp = lane >> 4;        // k-pair select

    // staging indices: A = 128x32 floats (4 x b128/thread), B = 32x128 (same)
    const int a_cg = tid & 7;        // 4-float col group in A slab
    const int a_r  = tid >> 3;       // 0..31
    const int b_ng = tid & 31;       // 4-float col group in B slab
    const int b_kr = tid >> 5;       // 0..7

    auto stage = [&](int kt, int buf) {
#if USE_ASYNC
#pragma unroll
        for (int it = 0; it < 4; ++it) {
            async_cp16(&Fz[(size_t)(bm * 128 + a_r + 32 * it) * MULC + kt * KT + a_cg * 4],
                       &As[buf][(a_r + 32 * it) * ASTR + a_cg * 4]);
            async_cp16(&Wt[(size_t)(kt * KT + b_kr + 8 * it) * MULC + bn * 128 + b_ng * 4],
                       &Bs[buf][(b_kr + 8 * it) * 128 + b_ng * 4]);
        }
#else
#pragma unroll
        for (int it = 0; it < 4; ++it) {
            *(float4*)&As[buf][(a_r + 32 * it) * ASTR + a_cg * 4] =
                *(const float4*)&Fz[(size_t)(bm * 128 + a_r + 32 * it) * MULC + kt * KT + a_cg * 4];
            *(float4*)&Bs[buf][(b_kr + 8 * it) * 128 + b_ng * 4] =
                *(const float4*)&Wt[(size_t)(kt * KT + b_kr + 8 * it) * MULC + bn * 128 + b_ng * 4];
        }
#endif
    };

#if USE_ASYNC
    stage(0, 0);                      // prologue: 8 async b128 per wave
#endif

    for (int kt = 0; kt < NKT; ++kt) {
#if USE_ASYNC
        const int cur = kt & 1;
        if (kt + 1 < NKT) {
            stage(kt + 1, cur ^ 1);                  // prefetch next slab
            __builtin_amdgcn_s_wait_asynccnt(8);     // slab kt landed (in-order)
        } else {
            __builtin_amdgcn_s_wait_asynccnt(0);     // drain
        }
        __syncthreads();
#else
        const int cur = 0;
        stage(kt, 0);
        __syncthreads();
#endif

#pragma unroll
        for (int ks = 0; ks < 8; ++ks) {
            const int kk = ks * 4 + 2 * kp;          // this lane's k pair
            v2f afr[4], bfr[2];
#pragma unroll
            for (int mt = 0; mt < 4; ++mt)
                afr[mt] = *(const v2f*)&As[cur][(wm * 64 + mt * 16 + ml) * ASTR + kk];
#pragma unroll
            for (int nt = 0; nt < 2; ++nt) {
                const int col = wn * 32 + nt * 16 + ml;
                v2f t = { Bs[cur][kk * 128 + col], Bs[cur][(kk + 1) * 128 + col] };
                bfr[nt] = t;
            }
#pragma unroll
            for (int mt = 0; mt < 4; ++mt)
#pragma unroll
                for (int nt = 0; nt < 2; ++nt)
                    acc[mt][nt] = __builtin_amdgcn_wmma_f32_16x16x4_f32(
                        false, afr[mt], false, bfr[nt],
                        (short)0, acc[mt][nt], false, false);
        }
        __syncthreads();
    }

    // epilogue: D reg r holds rows r (lanes 0-15) / r+8 (lanes 16-31), N = lane%16
    const int rofs = kp * 8;
#pragma unroll
    for (int mt = 0; mt < 4; ++mt) {
        int row_base = bm * 128 + wm * 64 + mt * 16 + rofs;
#pragma unroll
        for (int nt = 0; nt < 2; ++nt) {
            int col = bn * 128 + wn * 32 + nt * 16 + ml;
#pragma unroll
            for (int r = 0; r < 8; ++r)
                Gz[(size_t)(row_base + r) * MULC + col] = acc[mt][nt][r] * scale;
        }
    }
}

// ---------------------------------------------------------------------------
// out[b][i] = (1/32) * sum_v F[i][b][v] * w_out[v]
// ---------------------------------------------------------------------------
__global__ __launch_bounds__(256) void out_kernel(const float* __restrict__ F,
                                                  const float* __restrict__ w_out,
                                                  float* __restrict__ out) {
    __shared__ float red[256];
    int b   = blockIdx.x;
    int tid = threadIdx.x;
    for (int i = 0; i < 3; ++i) {
        float acc = 0.0f;
        for (int v = tid; v < MULC; v += 256)
            acc += F[(size_t)i * BV + (size_t)b * MULC + v] * w_out[v];
        red[tid] = acc;
        __syncthreads();
        for (int s = 128; s > 0; s >>= 1) {
            if (tid < s) red[tid] += red[tid + s];
            __syncthreads();
        }
        if (tid == 0) out[b * 3 + i] = red[0] * 0.03125f;
        __syncthreads();
    }
}

// ---------------------------------------------------------------------------
extern "C" void kernel_launch(void* const* d_in, const int* in_sizes, int n_in,
                              void* d_out, int out_size, void* d_ws, size_t ws_size,
                              hipStream_t stream) {
    (void)in_sizes; (void)n_in; (void)out_size; (void)ws_size;
    const float* x     = (const float*)d_in[0];   // [1024,2,3,1]
    const float* w1    = (const float*)d_in[1];   // [2,1024]
    const float* W     = (const float*)d_in[2];   // [5,4,1024,1024]
    const float* bn_w  = (const float*)d_in[3];   // [6,4,1024]
    const float* w_out = (const float*)d_in[4];   // [1024,1]
    float* out = (float*)d_out;                   // [1024,3]

    char* ws = (char*)d_ws;
    float* bufA = (float*)ws;                                    // [3][B][MUL] pre-BN
    float* bufB = (float*)(ws + (size_t)3 * BV * sizeof(float)); // [3][B][MUL] post-act
    float* var  = (float*)(ws + (size_t)6 * BV * sizeof(float)); // [MUL]

    // block 1 linear + bn_act
    block1_kernel<<<3 * BV / 256, 256, 0, stream>>>(x, w1, bufA);
    var_zero<<<MULC / 256, 256, 0, stream>>>(var);
    var_reduce<<<dim3(MULC / 256, 16), 256, 0, stream>>>(bufA, var);
    bn_act_kernel<<<BV / 256, 256, 0, stream>>>(bufA, var, bn_w + 0, bufB);

    // blocks 2..6: per-component GEMM (l=1 path only; l=2..4 are identically 0)
    for (int L = 0; L < 5; ++L) {
        const float* WL = W + (size_t)L * 4 * MULC * MULC;   // W[L][0]
        gemm_wmma_kernel<<<dim3(8, 8, 3), 256, 0, stream>>>(bufB, WL, bufA, 0.03125f);
        var_zero<<<MULC / 256, 256, 0, stream>>>(var);
        var_reduce<<<dim3(MULC / 256, 16), 256, 0, stream>>>(bufA, var);
        bn_act_kernel<<<BV / 256, 256, 0, stream>>>(bufA, var, bn_w + (size_t)(L + 1) * 4 * MULC, bufB);
    }

    // output projection
    out_kernel<<<BSZ, 256, 0, stream>>>(bufB, w_out, out);
}